// FeatureGCN_23158463660765
// MI455X (gfx1250) — compile-verified
//
#include <hip/hip_runtime.h>
#include <hip/hip_bf16.h>

// ---------------------------------------------------------------------------
// CDNA5 (gfx1250) GCN forward: f32 WMMA GEMMs + atomic edge aggregation.
// ---------------------------------------------------------------------------

typedef __attribute__((ext_vector_type(2))) float v2f;
typedef __attribute__((ext_vector_type(8))) float v8f;

#define NUM_GRAPHS_C 16
#define LN_EPS 1e-5f

// ---------------------------------------------------------------------------
// Degree / normalization kernels
// ---------------------------------------------------------------------------
__global__ void k_deg_init(float* __restrict__ deg, int n) {
    int i = blockIdx.x * blockDim.x + threadIdx.x;
    if (i < n) deg[i] = 1.0f;  // self-loop weight
}

__global__ void k_deg_edges(float* __restrict__ deg, const int* __restrict__ dst,
                            const float* __restrict__ ew, int e) {
    int i = blockIdx.x * blockDim.x + threadIdx.x;
    if (i < e) atomicAdd(&deg[dst[i]], ew[i]);
}

__global__ void k_dinv(float* __restrict__ deg, int n) {
    int i = blockIdx.x * blockDim.x + threadIdx.x;
    if (i < n) {
        float d = deg[i];
        deg[i] = (d > 0.0f) ? rsqrtf(d) : 0.0f;
    }
}

// ---------------------------------------------------------------------------
// f32 WMMA GEMM: C[M,N] = A[M,K] * B[K,N].
// One wave32 computes a 16x64 strip = 4 adjacent 16x16 tiles, so each A
// fragment load feeds 4 V_WMMA_F32_16X16X4_F32 issues (4 independent
// accumulator chains -> no WMMA->WMMA RAW hazards, good pipelining).
// M multiple of 16 (40000 = 2500*16); N multiple of 64; K multiple of 4.
// Fragment layout (ISA 7.12.2, 32-bit):
//   A 16x4 : lane l -> row l%16, VGPR0 = K(2*(l/16)),    VGPR1 = K(2*(l/16)+1)
//   B 4x16 : lane l -> col l%16, VGPR0 = K-row 2*(l/16), VGPR1 = next K-row
//   C 16x16: VGPR r, lane l -> row (l/16)*8 + r, col l%16
// ---------------------------------------------------------------------------
template <int N, int K>
__global__ __launch_bounds__(256) void k_gemm_wmma(const float* __restrict__ A,
                                                   const float* __restrict__ B,
                                                   float* __restrict__ C, int M) {
    const int lane = threadIdx.x & 31;
    const int wid  = threadIdx.x >> 5;
    constexpr int NG = N / 64;                 // groups of 4 n-tiles
    const int tile = blockIdx.x * 8 + wid;     // 8 waves per 256-thread block
    const int mt = tile / NG;
    const int ng = tile % NG;
    if (mt * 16 >= M) return;                  // wave-uniform exit (EXEC stays all-1)

    const int m0 = mt * 16;
    const int n0 = ng * 64;
    const int ml = lane & 15;
    const int hl = lane >> 4;                  // half-wave selector

    v8f acc0 = {}, acc1 = {}, acc2 = {}, acc3 = {};
    const float* __restrict__ Arow = A + (size_t)(m0 + ml) * K;
    const float* __restrict__ Bcol = B + n0 + ml;

    for (int k0 = 0; k0 < K; k0 += 4) {
        const int ka = k0 + 2 * hl;
        v2f a;
        a.x = Arow[ka];
        a.y = Arow[ka + 1];
        const float* __restrict__ Br0 = Bcol + (size_t)ka * N;
        const float* __restrict__ Br1 = Bcol + (size_t)(ka + 1) * N;
        v2f b0; b0.x = Br0[0];  b0.y = Br1[0];
        v2f b1; b1.x = Br0[16]; b1.y = Br1[16];
        v2f b2; b2.x = Br0[32]; b2.y = Br1[32];
        v2f b3; b3.x = Br0[48]; b3.y = Br1[48];
        acc0 = __builtin_amdgcn_wmma_f32_16x16x4_f32(false, a, false, b0, (short)0, acc0, false, false);
        acc1 = __builtin_amdgcn_wmma_f32_16x16x4_f32(false, a, false, b1, (short)0, acc1, false, false);
        acc2 = __builtin_amdgcn_wmma_f32_16x16x4_f32(false, a, false, b2, (short)0, acc2, false, false);
        acc3 = __builtin_amdgcn_wmma_f32_16x16x4_f32(false, a, false, b3, (short)0, acc3, false, false);
    }

    float* __restrict__ Crow = C + (size_t)(m0 + hl * 8) * N + n0 + ml;
#pragma unroll
    for (int r = 0; r < 8; ++r) {
        Crow[(size_t)r * N + 0]  = acc0[r];
        Crow[(size_t)r * N + 16] = acc1[r];
        Crow[(size_t)r * N + 32] = acc2[r];
        Crow[(size_t)r * N + 48] = acc3[r];
    }
}

// ---------------------------------------------------------------------------
// Aggregation: out[i] = bias + dinv[i]^2 * xw[i]  (self-loop term, no atomics)
// ---------------------------------------------------------------------------
template <int C>
__global__ void k_agg_init(const float* __restrict__ xw, const float* __restrict__ dinv,
                           const float* __restrict__ bias, float* __restrict__ out, int n) {
    const int idx = blockIdx.x * blockDim.x + threadIdx.x;   // over n * C/4
    const int total = n * (C / 4);
    if (idx >= total) return;
    const int node = idx / (C / 4);
    const int c4 = (idx % (C / 4)) * 4;
    const float di = dinv[node];
    const float s = di * di;
    const float4 v = *(const float4*)(xw + (size_t)node * C + c4);
    const float4 bb = *(const float4*)(bias + c4);
    float4 o;
    o.x = bb.x + s * v.x;
    o.y = bb.y + s * v.y;
    o.z = bb.z + s * v.z;
    o.w = bb.w + s * v.w;
    *(float4*)(out + (size_t)node * C + c4) = o;
}

// One block per edge; C/4 threads each scatter a float4 with f32 atomics.
template <int C>
__global__ void k_agg_edges(const float* __restrict__ xw, const int* __restrict__ src,
                            const int* __restrict__ dst, const float* __restrict__ ew,
                            const float* __restrict__ dinv, float* __restrict__ out, int E) {
    const int e = blockIdx.x;
    if (e >= E) return;
    const int s = src[e];
    const int d = dst[e];
    const float nrm = dinv[s] * ew[e] * dinv[d];
    const float* __restrict__ xs = xw + (size_t)s * C;
    float* __restrict__ od = out + (size_t)d * C;
    for (int c = threadIdx.x * 4; c < C; c += blockDim.x * 4) {
        const float4 v = *(const float4*)(xs + c);
        atomicAdd(od + c + 0, nrm * v.x);
        atomicAdd(od + c + 1, nrm * v.y);
        atomicAdd(od + c + 2, nrm * v.z);
        atomicAdd(od + c + 3, nrm * v.w);
    }
}

// ---------------------------------------------------------------------------
// Fused ReLU + LayerNorm, one 256-thread block per row, in place.
// ---------------------------------------------------------------------------
template <int C>
__global__ __launch_bounds__(256) void k_relu_ln(float* __restrict__ h,
                                                 const float* __restrict__ g,
                                                 const float* __restrict__ be) {
    constexpr int T = 256;
    constexpr int PER = C / T;                 // 2 (C=512) or 1 (C=256)
    __shared__ float red[T];
    const int tid = threadIdx.x;
    float* __restrict__ hp = h + (size_t)blockIdx.x * C;

    float v[PER];
    float s = 0.0f;
#pragma unroll
    for (int i = 0; i < PER; ++i) {
        float t = hp[tid + i * T];
        v[i] = t > 0.0f ? t : 0.0f;            // ReLU before LN (per reference)
        s += v[i];
    }
    red[tid] = s;
    __syncthreads();
    for (int off = T / 2; off > 0; off >>= 1) {
        if (tid < off) red[tid] += red[tid + off];
        __syncthreads();
    }
    const float mu = red[0] / (float)C;
    __syncthreads();

    float s2 = 0.0f;
#pragma unroll
    for (int i = 0; i < PER; ++i) {
        float d = v[i] - mu;
        s2 += d * d;
    }
    red[tid] = s2;
    __syncthreads();
    for (int off = T / 2; off > 0; off >>= 1) {
        if (tid < off) red[tid] += red[tid + off];
        __syncthreads();
    }
    const float var = red[0] / (float)C;
    const float inv = rsqrtf(var + LN_EPS);

#pragma unroll
    for (int i = 0; i < PER; ++i) {
        const int c = tid + i * T;
        hp[c] = (v[i] - mu) * inv * g[c] + be[c];
    }
}

// ---------------------------------------------------------------------------
// Global mean pool + final FC
// ---------------------------------------------------------------------------
__global__ void k_pool_zero(float* __restrict__ pool, float* __restrict__ cnt) {
    const int i = blockIdx.x * blockDim.x + threadIdx.x;
    if (i < NUM_GRAPHS_C * 256) pool[i] = 0.0f;
    if (i < NUM_GRAPHS_C) cnt[i] = 0.0f;
}

__global__ void k_pool(const float* __restrict__ h3, const int* __restrict__ batch,
                       float* __restrict__ pool, float* __restrict__ cnt, int n) {
    const int node = blockIdx.x;
    if (node >= n) return;
    const int gid = batch[node];
    atomicAdd(pool + (size_t)gid * 256 + threadIdx.x, h3[(size_t)node * 256 + threadIdx.x]);
    if (threadIdx.x == 0) atomicAdd(cnt + gid, 1.0f);
}

__global__ void k_fc(const float* __restrict__ pool, const float* __restrict__ cnt,
                     const float* __restrict__ Wfc, const float* __restrict__ bfc,
                     float* __restrict__ out) {
    const int i = threadIdx.x;                 // 32 outputs: 16 graphs x 2 channels
    if (i >= NUM_GRAPHS_C * 2) return;
    const int gidx = i >> 1;
    const int o = i & 1;
    float c = cnt[gidx];
    c = c > 1.0f ? c : 1.0f;
    const float invc = 1.0f / c;
    float acc = bfc[o];
    const float* __restrict__ pg = pool + (size_t)gidx * 256;
    for (int k = 0; k < 256; ++k) acc += (pg[k] * invc) * Wfc[k * 2 + o];
    out[gidx * 2 + o] = acc;
}

// ---------------------------------------------------------------------------
// Host-side launch (graph-capture safe: no allocs, no syncs)
// ---------------------------------------------------------------------------
extern "C" void kernel_launch(void* const* d_in, const int* in_sizes, int n_in,
                              void* d_out, int out_size, void* d_ws, size_t ws_size,
                              hipStream_t stream) {
    const float* x   = (const float*)d_in[0];
    const int*   ei  = (const int*)d_in[1];
    const float* ew  = (const float*)d_in[2];
    const int*   bat = (const int*)d_in[3];
    const float* W1  = (const float*)d_in[4];
    const float* b1  = (const float*)d_in[5];
    const float* g1  = (const float*)d_in[6];
    const float* be1 = (const float*)d_in[7];
    const float* W2  = (const float*)d_in[8];
    const float* b2  = (const float*)d_in[9];
    const float* g2  = (const float*)d_in[10];
    const float* be2 = (const float*)d_in[11];
    const float* Wfc = (const float*)d_in[12];
    const float* bfc = (const float*)d_in[13];

    const int N = in_sizes[0] / 128;   // 40000 nodes
    const int E = in_sizes[1] / 2;     // 640000 edges
    const int* srcI = ei;              // edge_index[0]
    const int* dstI = ei + E;          // edge_index[1]

    // Workspace layout (floats):
    //   xw   : N*512   (GEMM output; reused for layer 2's N*256 output)
    //   hbuf : N*512   (layer-1 hidden after aggregation/LN)
    //   dinv : N       (degree -> deg^{-1/2}, computed in place)
    //   pool : 16*256, cnt : 16
    float* ws   = (float*)d_ws;
    float* xw   = ws;
    float* hbuf = ws + (size_t)N * 512;
    float* dinv = hbuf + (size_t)N * 512;
    float* pool = dinv + N;
    float* cnt  = pool + NUM_GRAPHS_C * 256;

    float* h3   = (float*)d_out;                 // [N,256] first output
    float* outv = h3 + (size_t)N * 256;          // [16,2] second output

    // --- degrees -> deg^{-1/2} -------------------------------------------
    k_deg_init<<<(N + 255) / 256, 256, 0, stream>>>(dinv, N);
    k_deg_edges<<<(E + 255) / 256, 256, 0, stream>>>(dinv, dstI, ew, E);
    k_dinv<<<(N + 255) / 256, 256, 0, stream>>>(dinv, N);

    // --- layer 1: xw = x @ W1 ; aggregate ; ReLU+LN -----------------------
    {
        const int waves = (N / 16) * (512 / 64);   // 16x64 strip per wave
        k_gemm_wmma<512, 128><<<(waves + 7) / 8, 256, 0, stream>>>(x, W1, xw, N);
    }
    k_agg_init<512><<<((N * 128) + 255) / 256, 256, 0, stream>>>(xw, dinv, b1, hbuf, N);
    k_agg_edges<512><<<E, 128, 0, stream>>>(xw, srcI, dstI, ew, dinv, hbuf, E);
    k_relu_ln<512><<<N, 256, 0, stream>>>(hbuf, g1, be1);

    // --- layer 2: xw2 = h @ W2 ; aggregate into d_out ; ReLU+LN -----------
    {
        const int waves = (N / 16) * (256 / 64);
        k_gemm_wmma<256, 512><<<(waves + 7) / 8, 256, 0, stream>>>(hbuf, W2, xw, N);
    }
    k_agg_init<256><<<((N * 64) + 255) / 256, 256, 0, stream>>>(xw, dinv, b2, h3, N);
    k_agg_edges<256><<<E, 64, 0, stream>>>(xw, srcI, dstI, ew, dinv, h3, E);
    k_relu_ln<256><<<N, 256, 0, stream>>>(h3, g2, be2);

    // --- mean pool + FC ---------------------------------------------------
    k_pool_zero<<<(NUM_GRAPHS_C * 256 + 255) / 256, 256, 0, stream>>>(pool, cnt);
    k_pool<<<N, 256, 0, stream>>>(h3, bat, pool, cnt, N);
    k_fc<<<1, 32, 0, stream>>>(pool, cnt, Wfc, bfc, outv);
}